// SPL_ComputeWithTrace_4818953306340
// MI455X (gfx1250) — compile-verified
//
#include <hip/hip_runtime.h>

// Reference: fused cosine-similarity trace over [16,3,512,512] float32.
// Memory-bound: ~100.7 MB reads -> ~4.3 us at 23.3 TB/s. Two self-contained
// passes (rows, then columns); second pass is served from the 192 MB L2.
// Per-row/column sum reductions are done on the matrix pipe via
// V_WMMA_F32_16X16X4_F32 with an all-ones B: C += A x 1 gives row-sums of A
// using only the documented A (16x4 f32) and C (16x16 f32) VGPR layouts.

#define EPSN 1e-12f

typedef __attribute__((ext_vector_type(2))) float v2f;
typedef __attribute__((ext_vector_type(8))) float v8f;

template <bool COL>
__global__ __launch_bounds__(256) void sim_pass(const float* __restrict__ x,
                                                const float* __restrict__ y,
                                                const float* __restrict__ wgt,
                                                float* __restrict__ accum) {
  constexpr int Hd = 512, Wd = 512;
  const int s    = blockIdx.x >> 5;  // slice 0..47 (b*3+c)
  const int t    = blockIdx.x & 31;  // strip of 16 rows (row pass) or 16 cols (col pass)
  const int tid  = threadIdx.x;
  const int wave = tid >> 5;         // 8 wave32s per block
  const int lane = tid & 31;
  const int m    = lane & 15;        // M index (row or column within strip)
  const int kp   = (lane >> 4) << 1; // K sub-offset: lanes 0-15 -> K 0,1; lanes 16-31 -> K 2,3

  const int base = s * Hd * Wd;

  v8f cx2 = {}; // per-M running sums of x*x (replicated across N)
  v8f cy2 = {}; // per-M running sums of y*y
  v8f cxy = {}; // per-M running sums of x*y
  v2f ones; ones.x = 1.0f; ones.y = 1.0f; // all-ones B: layout-invariant

  // Each wave reduces a 64-element span of the K axis in chunks of 4.
  const int k0 = wave * 64;
#pragma unroll 4
  for (int c0 = k0; c0 < k0 + 64; c0 += 4) {
    float x0, x1, y0, y1;
    if (!COL) {
      // Row pass: M = row (t*16+m), K = column. K pair is contiguous -> float2 loads.
      const int off = base + (t * 16 + m) * Wd + (c0 + kp);
      v2f vx = *(const v2f*)(x + off);
      v2f vy = *(const v2f*)(y + off);
      x0 = vx.x; x1 = vx.y; y0 = vy.x; y1 = vy.y;
    } else {
      // Col pass: M = column (t*16+m), K = row. Lanes 0-15 read 64 contiguous
      // bytes of one row, lanes 16-31 of another -> coalesced b32 loads.
      const int off = base + (c0 + kp) * Wd + (t * 16 + m);
      x0 = x[off]; x1 = x[off + Wd];
      y0 = y[off]; y1 = y[off + Wd];
    }
    v2f ax2, ay2, axy;
    ax2.x = x0 * x0; ax2.y = x1 * x1;
    ay2.x = y0 * y0; ay2.y = y1 * y1;
    axy.x = x0 * y0; axy.y = x1 * y1;
    // D = A x ones + C : accumulates per-M sums on the matrix pipe.
    cx2 = __builtin_amdgcn_wmma_f32_16x16x4_f32(false, ax2, false, ones, (short)0, cx2, false, false);
    cy2 = __builtin_amdgcn_wmma_f32_16x16x4_f32(false, ay2, false, ones, (short)0, cy2, false, false);
    cxy = __builtin_amdgcn_wmma_f32_16x16x4_f32(false, axy, false, ones, (short)0, cxy, false, false);
  }

  // C layout: lane 0 holds M=0..7 in v0..7 (N=0), lane 16 holds M=8..15.
  __shared__ float acc[16][3];
  __shared__ float terms[16];
  if (tid < 48) (&acc[0][0])[tid] = 0.0f;
  __syncthreads();
  if ((lane & 15) == 0) {
    const int mb = (lane >> 4) * 8;
#pragma unroll
    for (int r = 0; r < 8; ++r) {
      atomicAdd(&acc[mb + r][0], cx2[r]);
      atomicAdd(&acc[mb + r][1], cy2[r]);
      atomicAdd(&acc[mb + r][2], cxy[r]);
    }
  }
  __syncthreads();
  if (tid < 16) {
    const float sx  = acc[tid][0];
    const float sy  = acc[tid][1];
    const float sxy = acc[tid][2];
    terms[tid] = sxy / (fmaxf(sqrtf(sx), EPSN) * fmaxf(sqrtf(sy), EPSN));
  }
  __syncthreads();
  if (tid == 0) {
    float sum = 0.0f;
#pragma unroll
    for (int i = 0; i < 16; ++i) sum += terms[i];
    atomicAdd(accum, sum * wgt[s % 3]); // weight by channel c = s % 3
  }
}

__global__ void sim_finalize(const float* __restrict__ acc, float* __restrict__ out) {
  // out = -( R/H + C/W ) / B  with H = W = 512, B = 16
  out[0] = -(acc[0] + acc[1]) * (1.0f / (512.0f * 16.0f));
}

extern "C" void kernel_launch(void* const* d_in, const int* in_sizes, int n_in,
                              void* d_out, int out_size, void* d_ws, size_t ws_size,
                              hipStream_t stream) {
  const float* x = (const float*)d_in[0];
  const float* y = (const float*)d_in[1];
  const float* w = (const float*)d_in[2];
  float* accum   = (float*)d_ws; // accum[0] = row-pass sum, accum[1] = col-pass sum

  hipMemsetAsync(d_ws, 0, 2 * sizeof(float), stream); // capture-safe memset node

  dim3 grid(48 * 32);
  dim3 block(256);
  sim_pass<false><<<grid, block, 0, stream>>>(x, y, w, accum + 0);
  sim_pass<true ><<<grid, block, 0, stream>>>(x, y, w, accum + 1);
  sim_finalize<<<1, 1, 0, stream>>>(accum, (float*)d_out);
}